// AEVComputerJoint_68289980006640
// MI455X (gfx1250) — compile-verified
//
#include <hip/hip_runtime.h>
#include <hip/hip_bf16.h>
#include <math.h>

typedef __attribute__((ext_vector_type(2))) float v2f;
typedef __attribute__((ext_vector_type(8))) float v8f;

#define RCR_ 5.2f
#define RCA_ 3.5f
#define PI_  3.14159265358979323846f
#define NATOMS 48
#define NCHUNKS 282   /* 1128 pairs / 4 per WMMA step */

union V8 { v8f v; float f[8]; };

__device__ __forceinline__ int pair_index4(int s1, int s2) {
    int smin = s1 < s2 ? s1 : s2;
    int smax = s1 < s2 ? s2 : s1;
    // TRIU index for upper-triangular pair table, N=4
    return smin * 4 - (smin * (smin - 1)) / 2 + (smax - smin);
}

__global__ __launch_bounds__(128) void aev_kernel(
    const float* __restrict__ coords,   // (C, 48, 3)
    const float* __restrict__ EtaR,
    const float* __restrict__ ShfR,     // 16
    const float* __restrict__ EtaA,
    const float* __restrict__ Zeta,
    const float* __restrict__ ShfA,     // 4
    const float* __restrict__ ShfZ,     // 8
    const int*   __restrict__ species,  // (C, 48)
    float*       __restrict__ out)      // (C, 48, 384)
{
    const int b    = blockIdx.x;
    const int c    = b / NATOMS;
    const int i    = b % NATOMS;
    const int tid  = threadIdx.x;
    const int lane = tid & 31;
    const int wave = tid >> 5;      // 0..3
    const int hi   = lane >> 4;     // which half-wave (K split)
    const int n    = lane & 15;     // N (and M) index within tile
    const int kkLo = hi * 2;        // K base held by this half-wave

    __shared__ float sx[NATOMS], sy[NATOMS], sz[NATOMS];
    __shared__ float sd[NATOMS], sfcA[NATOMS], sfcR[NATOMS];
    __shared__ int   ssp[NATOMS];
    __shared__ float angacc[320];

    // zero angular accumulator
    for (int idx = tid; idx < 320; idx += 128) angacc[idx] = 0.0f;

    // per-neighbor geometry precompute
    if (tid < NATOMS) {
        int j = tid;
        const float* cb = coords + ((size_t)c * NATOMS) * 3;
        float dx = cb[j * 3 + 0] - cb[i * 3 + 0];
        float dy = cb[j * 3 + 1] - cb[i * 3 + 1];
        float dz = cb[j * 3 + 2] - cb[i * 3 + 2];
        float d  = sqrtf(dx * dx + dy * dy + dz * dz);
        sx[j] = dx; sy[j] = dy; sz[j] = dz; sd[j] = d;
        bool ok = (j != i);
        sfcR[j] = (ok && d <= RCR_) ? (0.5f * cosf(d * (PI_ / RCR_)) + 0.5f) : 0.0f;
        sfcA[j] = (ok && d <= RCA_) ? (0.5f * cosf(d * (PI_ / RCA_)) + 0.5f) : 0.0f;
        ssp[j] = species[c * NATOMS + j];
    }
    __syncthreads();

    const float etaR = EtaR[0], etaA = EtaA[0], zeta = Zeta[0];
    float* outA = out + ((size_t)c * NATOMS + i) * 384;

    // ---------------- radial: D(16x16) = onehot(16x4-chunks)^ * rad, K over 48 j ----------------
    if (wave == 0) {
        float shfr = ShfR[n];
        V8 accR;
        #pragma unroll
        for (int v = 0; v < 8; ++v) accR.f[v] = 0.0f;
        for (int cc = 0; cc < 12; ++cc) {
            int jA = cc * 4 + kkLo;
            int jB = jA + 1;
            v2f Am, Bm;
            Am.x = (ssp[jA] == n) ? 1.0f : 0.0f;   // A[M=n][K=kkLo]
            Am.y = (ssp[jB] == n) ? 1.0f : 0.0f;   // A[M=n][K=kkLo+1]
            float tA = sd[jA] - shfr;
            float tB = sd[jB] - shfr;
            Bm.x = 0.25f * expf(-etaR * tA * tA) * sfcR[jA];  // B[K=kkLo][N=n]
            Bm.y = 0.25f * expf(-etaR * tB * tB) * sfcR[jB];  // B[K=kkLo+1][N=n]
            accR.v = __builtin_amdgcn_wmma_f32_16x16x4_f32(
                false, Am, false, Bm, (short)0, accR.v, false, false);
        }
        if (hi == 0) {  // rows M=0..7 live in lanes 0..15; only M<4 (species) valid
            #pragma unroll
            for (int v = 0; v < 4; ++v) outA[v * 16 + n] = accR.f[v];
        }
    }

    // ---------------- angular: two 16-wide feature tiles, K over 1128 pairs ----------------
    const int   z     = n & 7;
    const float zshf  = ShfZ[z];
    const float shfA0 = ShfA[n >> 3];        // a in {0,1} for tile0
    const float shfA1 = ShfA[(n >> 3) + 2];  // a in {2,3} for tile1
    V8 a0, a1;
    #pragma unroll
    for (int v = 0; v < 8; ++v) { a0.f[v] = 0.0f; a1.f[v] = 0.0f; }

    for (int chunk = wave; chunk < NCHUNKS; chunk += 4) {
        int base = chunk * 4;
        float b00 = 0.0f, b01 = 0.0f, b10 = 0.0f, b11 = 0.0f;
        int p0 = 0, p1 = 0;
        #pragma unroll
        for (int q = 0; q < 2; ++q) {
            int t = base + kkLo + q;  // pair id, 0..1127, enumerates (j<k)
            // decode j,k from triangular index: rowStart(j) = j*(95-j)/2
            int j = (int)floorf((95.0f - sqrtf(fmaxf(9025.0f - 8.0f * (float)t, 0.0f))) * 0.5f);
            if (j < 0) j = 0;
            if (j > 46) j = 46;
            while (j > 0 && (j * (95 - j)) / 2 > t) --j;
            while (j < 46 && ((j + 1) * (94 - j)) / 2 <= t) ++j;
            int k = t - (j * (95 - j)) / 2 + j + 1;

            float dj = sd[j], dk = sd[k];
            float dot = sx[j] * sx[k] + sy[j] * sy[k] + sz[j] * sz[k];
            float cosang = dot / fmaxf(dj * dk, 1e-10f);
            float theta = acosf(0.95f * cosang);
            float fc12 = sfcA[j] * sfcA[k];           // 0 when pair masked
            float avg = 0.5f * (dj + dk);
            float f1 = powf(0.5f * (1.0f + cosf(theta - zshf)), zeta);
            float e0 = avg - shfA0, e1 = avg - shfA1;
            float cm = 2.0f * f1 * fc12;
            float v0 = cm * expf(-etaA * e0 * e0);    // feature (a_lo, z)
            float v1 = cm * expf(-etaA * e1 * e1);    // feature (a_lo+2, z)
            int p = pair_index4(ssp[j], ssp[k]);
            if (q == 0) { b00 = v0; b01 = v1; p0 = p; }
            else        { b10 = v0; b11 = v1; p1 = p; }
        }
        v2f Am, B0, B1;
        Am.x = (p0 == n) ? 1.0f : 0.0f;  // one-hot over pair type, M=n
        Am.y = (p1 == n) ? 1.0f : 0.0f;
        B0.x = b00; B0.y = b10;          // tile0: features 0..15
        B1.x = b01; B1.y = b11;          // tile1: features 16..31
        a0.v = __builtin_amdgcn_wmma_f32_16x16x4_f32(
            false, Am, false, B0, (short)0, a0.v, false, false);
        a1.v = __builtin_amdgcn_wmma_f32_16x16x4_f32(
            false, Am, false, B1, (short)0, a1.v, false, false);
    }

    // cross-wave reduction of D tiles into LDS (rows M<10 are valid pair types)
    #pragma unroll
    for (int v = 0; v < 8; ++v) {
        int M = v + hi * 8;
        if (M < 10) {
            atomicAdd(&angacc[M * 32 + n],       a0.f[v]);
            atomicAdd(&angacc[M * 32 + 16 + n],  a1.f[v]);
        }
    }
    __syncthreads();

    for (int idx = tid; idx < 320; idx += 128) outA[64 + idx] = angacc[idx];
}

extern "C" void kernel_launch(void* const* d_in, const int* in_sizes, int n_in,
                              void* d_out, int out_size, void* d_ws, size_t ws_size,
                              hipStream_t stream) {
    (void)in_sizes; (void)n_in; (void)out_size; (void)d_ws; (void)ws_size;
    const float* coords  = (const float*)d_in[0];
    const float* EtaR    = (const float*)d_in[1];
    const float* ShfR    = (const float*)d_in[2];
    const float* EtaA    = (const float*)d_in[3];
    const float* Zeta    = (const float*)d_in[4];
    const float* ShfA    = (const float*)d_in[5];
    const float* ShfZ    = (const float*)d_in[6];
    const int*   spec    = (const int*)d_in[7];
    float* out = (float*)d_out;

    dim3 grid(32 * NATOMS);   // one workgroup per (conformer, atom)
    dim3 block(128);          // 4 wave32
    hipLaunchKernelGGL(aev_kernel, grid, block, 0, stream,
                       coords, EtaR, ShfR, EtaA, Zeta, ShfA, ShfZ, spec, out);
}